// overlap_loss_11441792876694
// MI455X (gfx1250) — compile-verified
//
#include <hip/hip_runtime.h>

typedef float v2f __attribute__((ext_vector_type(2)));
typedef float v8f __attribute__((ext_vector_type(8)));

#define HW      (640 * 640)          // 409600
#define BATCH   8
#define NINST   16
#define NSEG    (BATCH * NINST)      // 128
#define TOT     (BATCH * HW)         // 3276800
#define TOT4    (TOT / 4)            // 819200
#define HW4     (HW / 4)             // 102400
#define WS_SEG  0                    // ws[0..127]   = cnt
                                     // ws[128..255] = s1
                                     // ws[256..383] = s2
#define WS_SCAL (3 * NSEG)           // ws[384..387] = S_or, S_pos, N_pos, S_neg
#define WS_LEN  512

__device__ __forceinline__ float bce_f(float p, float t) {
    float lp  = fmaxf(logf(p),     -100.0f);
    float l1p = fmaxf(log1pf(-p),  -100.0f);
    return -(t * lp + (1.0f - t) * l1p);
}

// Full wave32 sum using one V_WMMA_F32_16X16X4_F32.
// A-tile 16x4 f32: lanes 0-15 supply (M=lane, K=0) in VGPR0, lanes 16-31 supply
// (M=lane-16, K=2); VGPR1 (K=1/K=3) set to 0.  B = all ones (4x16).
// => D[m,n] = v[m] + v[m+16].  D VGPR j holds row j (lanes 0-15) / row j+8
// (lanes 16-31), so the local sum of the 8 D regs gives rows 0-7 on the low
// half-wave and rows 8-15 on the high half-wave; one shfl_xor(16) finishes.
__device__ __forceinline__ float wave_sum32(float v) {
    v2f a; a[0] = v;    a[1] = 0.0f;
    v2f b; b[0] = 1.0f; b[1] = 1.0f;
    v8f c = {};
    v8f d = __builtin_amdgcn_wmma_f32_16x16x4_f32(
        /*neg_a=*/false, a, /*neg_b=*/false, b,
        /*c_mod=*/(short)0, c, /*reuse_a=*/false, /*reuse_b=*/false);
    float s = d[0] + d[1] + d[2] + d[3] + d[4] + d[5] + d[6] + d[7];
    s += __shfl_xor(s, 16, 32);
    return s;
}

__global__ void __launch_bounds__(512)
ws_init_kernel(float* __restrict__ ws) {
    int t = threadIdx.x;
    if (t < WS_LEN) ws[t] = 0.0f;
}

__global__ void __launch_bounds__(256)
overlap_loss_main_kernel(const float* __restrict__ preds,
                         const float* __restrict__ conf,
                         const int*   __restrict__ inst,
                         const float* __restrict__ overlap,
                         float*       __restrict__ ws) {
    // Wave-private segment tables: no inter-wave ds_add contention.
    __shared__ float lseg[8][3 * NSEG];          // 12 KB
    const int lane = threadIdx.x & 31;
    const int wave = threadIdx.x >> 5;
    float* myseg = lseg[wave];
    for (int i = lane; i < 3 * NSEG; i += 32) myseg[i] = 0.0f;
    // LDS ops are in-order within a wave; no cross-wave dependency here.

    float s_or = 0.0f, s_pos = 0.0f, n_pos = 0.0f, s_neg = 0.0f;

    const int stride = blockDim.x * gridDim.x;
    for (int q = blockIdx.x * blockDim.x + threadIdx.x; q < TOT4; q += stride) {
        const int b   = q / HW4;
        const int hw4 = q - b * HW4;
        const float* pbase = preds + (size_t)b * (2 * HW) + (size_t)hw4 * 4;

        const float4 P1 = *(const float4*)(pbase);
        const float4 P2 = *(const float4*)(pbase + HW);
        const float4 CF = ((const float4*)conf)[q];
        const float4 OV = ((const float4*)overlap)[q];
        const int4   IT = ((const int4*)inst)[q];

        const float p1v[4] = {P1.x, P1.y, P1.z, P1.w};
        const float p2v[4] = {P2.x, P2.y, P2.z, P2.w};
        const float cfv[4] = {CF.x, CF.y, CF.z, CF.w};
        const float ovv[4] = {OV.x, OV.y, OV.z, OV.w};
        const int   itv[4] = {IT.x, IT.y, IT.z, IT.w};

#pragma unroll
        for (int j = 0; j < 4; ++j) {
            const float p1 = p1v[j], p2 = p2v[j];
            const float cf = cfv[j], ov = ovv[j];
            const float ap = p1 * p2;

            const float al = bce_f(ap, ov);                 // and_loss_map
            s_or += bce_f(fmaxf(p1, p2), cf);               // or bce

            const float isp = (cf == 1.0f) ? 1.0f : 0.0f;   // conf in {0,1}
            s_pos += isp * al;
            n_pos += isp;
            s_neg += (1.0f - isp) * al;

            const float ao = ap * ov;
            const float o1 = fmaxf(p1, ao) - 1.0f;
            const float o2 = fmaxf(p2, ao) - 1.0f;

            const int seg = (b << 4) + (itv[j] & (NINST - 1));
            atomicAdd(&myseg[seg],            1.0f);        // ds_add_f32
            atomicAdd(&myseg[NSEG + seg],     o1 * o1);
            atomicAdd(&myseg[2 * NSEG + seg], o2 * o2);
        }
    }
    __syncthreads();

    // Flush block's 8 wave tables -> global (one f32 atomic per entry).
    for (int i = threadIdx.x; i < 3 * NSEG; i += blockDim.x) {
        float s = 0.0f;
#pragma unroll
        for (int w = 0; w < 8; ++w) s += lseg[w][i];
        unsafeAtomicAdd(&ws[i], s);
    }

    // Scalar accumulators: WMMA-based wave32 reduction (EXEC all-1s here).
    const float a0 = wave_sum32(s_or);
    const float a1 = wave_sum32(s_pos);
    const float a2 = wave_sum32(n_pos);
    const float a3 = wave_sum32(s_neg);
    if (lane == 0) {
        unsafeAtomicAdd(&ws[WS_SCAL + 0], a0);
        unsafeAtomicAdd(&ws[WS_SCAL + 1], a1);
        unsafeAtomicAdd(&ws[WS_SCAL + 2], a2);
        unsafeAtomicAdd(&ws[WS_SCAL + 3], a3);
    }
}

__global__ void __launch_bounds__(128)
overlap_loss_finalize_kernel(const float* __restrict__ ws,
                             float* __restrict__ out) {
    __shared__ float perb[BATCH];
    const int t = threadIdx.x;                   // 0..127 == one segment each

    const float cnt = ws[t];
    const float s1  = ws[NSEG + t];
    const float s2  = ws[2 * NSEG + t];

    float pk = 0.0f, nk = 0.0f;
    if (cnt > 0.0f) {
        pk = 1.0f - fabsf(s1 / cnt - s2 / cnt);
        nk = 1.0f;
    }
    // Reduce across the 16 instances of each batch (16-lane groups in-wave).
#pragma unroll
    for (int off = 1; off < 16; off <<= 1) {
        pk += __shfl_xor(pk, off, 32);
        nk += __shfl_xor(nk, off, 32);
    }
    if ((t & 15) == 0) perb[t >> 4] = pk / nk;
    __syncthreads();

    if (t == 0) {
        float il = 0.0f;
#pragma unroll
        for (int b = 0; b < BATCH; ++b) il += perb[b];
        il *= (1.0f / (float)BATCH);

        const float S_or  = ws[WS_SCAL + 0];
        const float S_pos = ws[WS_SCAL + 1];
        const float N_pos = ws[WS_SCAL + 2];
        const float S_neg = ws[WS_SCAL + 3];
        const float N_neg = (float)TOT - N_pos;

        const float or_loss  = S_or / (float)TOT;
        const float and_loss = S_pos / N_pos + S_neg / N_neg;
        out[0] = 0.5f * and_loss + 0.25f * or_loss + 0.25f * il;
    }
}

extern "C" void kernel_launch(void* const* d_in, const int* in_sizes, int n_in,
                              void* d_out, int out_size, void* d_ws, size_t ws_size,
                              hipStream_t stream) {
    (void)in_sizes; (void)n_in; (void)out_size; (void)ws_size;
    const float* preds   = (const float*)d_in[0];
    const float* conf    = (const float*)d_in[1];
    const int*   inst    = (const int*)  d_in[2];
    const float* overlap = (const float*)d_in[3];
    float*       ws      = (float*)d_ws;
    float*       out     = (float*)d_out;

    ws_init_kernel<<<1, 512, 0, stream>>>(ws);
    // 800 blocks x 256 threads: each thread handles exactly 4 float4 groups.
    overlap_loss_main_kernel<<<800, 256, 0, stream>>>(preds, conf, inst, overlap, ws);
    overlap_loss_finalize_kernel<<<1, 128, 0, stream>>>(ws, out);
}